// ProdLDAEncoder_52372831207607
// MI455X (gfx1250) — compile-verified
//
#include <hip/hip_runtime.h>
#include <math.h>

typedef __attribute__((ext_vector_type(2))) float v2f;
typedef __attribute__((ext_vector_type(8))) float v8f;

// ---------------------------------------------------------------------------
// degree / normalization
// ---------------------------------------------------------------------------
__global__ void k_fill(float* __restrict__ p, float v, unsigned n) {
  unsigned i = blockIdx.x * blockDim.x + threadIdx.x;
  if (i < n) p[i] = v;
}

__global__ void k_count_deg(const long long* __restrict__ dst,
                            float* __restrict__ deg, unsigned E) {
  unsigned i = blockIdx.x * blockDim.x + threadIdx.x;
  if (i < E) atomicAdd(&deg[dst[i]], 1.0f);
}

__global__ void k_rsqrt_ip(float* __restrict__ p, unsigned n) {
  unsigned i = blockIdx.x * blockDim.x + threadIdx.x;
  if (i < n) p[i] = rsqrtf(p[i]);   // deg >= 1 always (self loop)
}

// ---------------------------------------------------------------------------
// FP32 WMMA GEMM:  C[n x f_out] = A[n x f_in] * W[f_in x f_out]
// One block (8 wave32) owns a 16-row strip and ALL f_out columns.
// A strip (16 x f_in, <=32KB) staged once in LDS; each wave computes two
// adjacent 16x16 C tiles so one LDS A-fragment feeds two V_WMMA issues.
// A frag (16x4): lane(0-15)=M holds K-half 0..1, lanes 16-31 K-half 2..3;
//                vgpr = K%2.
// B frag (4x16): mirrored (lane = N, halves split K).
// C layout: vgpr r -> (M=r, lanes 0-15) / (M=r+8, lanes 16-31), N = lane&15.
// requires n % 16 == 0, f_in % 4 == 0, f_out % 32 == 0.
// ---------------------------------------------------------------------------
__global__ __launch_bounds__(256)
void k_gemm_wmma_f32(const float* __restrict__ A, const float* __restrict__ W,
                     float* __restrict__ C, int f_in, int f_out) {
  extern __shared__ float As[];              // 16 * f_in floats
  const int tid  = threadIdx.x;
  const int lane = tid & 31;
  const int wave = tid >> 5;
  const int tm   = blockIdx.x;               // 16-row strip

  // cooperative stage of the A strip (coalesced per row)
  const float* Ablk = A + (size_t)tm * 16 * f_in;
#pragma unroll
  for (int r = 0; r < 16; ++r)
    for (int c = tid; c < f_in; c += 256)
      As[r * f_in + c] = Ablk[(size_t)r * f_in + c];
  __syncthreads();

  const int tn0 = wave * 2;                  // first of two 16-col tiles
  if (tn0 * 16 >= f_out) return;

  const int half = lane >> 4;                // 0: K 0..1, 1: K 2..3
  const int mloc = lane & 15;
  const float* arow  = As + mloc * f_in + 2 * half;
  const int col0     = tn0 * 16 + mloc;
  const int col1     = col0 + 16;
  const float* wbase = W + (size_t)(2 * half) * f_out;

  v8f acc0 = {}, acc1 = {};
#pragma unroll 4
  for (int k = 0; k < f_in; k += 4) {
    v2f a, b0, b1;
    a.x = arow[k];                           // ds_load_b64
    a.y = arow[k + 1];
    const float* w0 = wbase + (size_t)k * f_out;
    b0.x = w0[col0];
    b0.y = w0[f_out + col0];
    b1.x = w0[col1];
    b1.y = w0[f_out + col1];
    acc0 = __builtin_amdgcn_wmma_f32_16x16x4_f32(false, a, false, b0,
                                                 (short)0, acc0, false, false);
    acc1 = __builtin_amdgcn_wmma_f32_16x16x4_f32(false, a, false, b1,
                                                 (short)0, acc1, false, false);
  }

  const size_t m0 = (size_t)tm * 16 + half * 8;
#pragma unroll
  for (int r = 0; r < 8; ++r) {
    C[(m0 + r) * (size_t)f_out + col0] = acc0[r];
    C[(m0 + r) * (size_t)f_out + col1] = acc1[r];
  }
}

// ---------------------------------------------------------------------------
// GCN aggregation: out = bias + self-loop + scatter-add over edges
// ---------------------------------------------------------------------------
__global__ void k_self_init(const float* __restrict__ hw,
                            const float* __restrict__ dinv,
                            const float* __restrict__ bias,
                            float* __restrict__ out,
                            unsigned n, int ld_hw, int F) {
  unsigned gid = blockIdx.x * blockDim.x + threadIdx.x;
  if (gid >= n * (unsigned)F) return;
  unsigned i = gid / (unsigned)F;
  int j = (int)(gid - i * (unsigned)F);
  float di = dinv[i];
  out[(size_t)i * F + j] = bias[j] + hw[(size_t)i * ld_hw + j] * di * di;
}

// F == 256 fast path: blockDim==256 -> block b handles edge b, thread = feature
__global__ __launch_bounds__(256)
void k_edge_scatter256(const float* __restrict__ hw,
                       const float* __restrict__ dinv,
                       const long long* __restrict__ src,
                       const long long* __restrict__ dst,
                       float* __restrict__ out, unsigned E) {
  unsigned e = blockIdx.x;
  if (e >= E) return;
  int j = threadIdx.x;
  long long s = src[e], d = dst[e];
  float w = dinv[s] * dinv[d];
  atomicAdd(&out[(size_t)d * 256 + j], hw[(size_t)s * 256 + j] * w);
}

__global__ void k_edge_scatter(const float* __restrict__ hw,
                               const float* __restrict__ dinv,
                               const long long* __restrict__ src,
                               const long long* __restrict__ dst,
                               float* __restrict__ out,
                               unsigned E, int ld_hw, int F) {
  unsigned gid = blockIdx.x * blockDim.x + threadIdx.x;
  if (gid >= E * (unsigned)F) return;
  unsigned e = gid / (unsigned)F;
  int j = (int)(gid - e * (unsigned)F);
  long long s = src[e], d = dst[e];
  float w = dinv[s] * dinv[d];
  atomicAdd(&out[(size_t)d * F + j], hw[(size_t)s * ld_hw + j] * w);
}

__global__ void k_softplus_ip(float* __restrict__ p, unsigned n) {
  unsigned i = blockIdx.x * blockDim.x + threadIdx.x;
  if (i >= n) return;
  float v = p[i];
  p[i] = fmaxf(v, 0.0f) + log1pf(expf(-fabsf(v)));
}

// zero-pad W[rows x K] -> Wp[rows x Kp]
__global__ void k_pad_w(const float* __restrict__ W, float* __restrict__ Wp,
                        int rows, int K, int Kp) {
  unsigned gid = blockIdx.x * blockDim.x + threadIdx.x;
  if (gid >= (unsigned)(rows * Kp)) return;
  int r = gid / Kp, k = gid - r * Kp;
  Wp[gid] = (k < K) ? W[r * K + k] : 0.0f;
}

// ---------------------------------------------------------------------------
// z = mu + sqrt(exp(logvar))*eps ; p = softmax(z) ; var = exp(logvar)
// one wave32 per node; lane covers k and k+32.
// out layout: z | p | mu | logvar | var  (each N*K)
// ---------------------------------------------------------------------------
__global__ void k_finalize(const float* __restrict__ eps,
                           float* __restrict__ out, unsigned n, int K) {
  int lane = threadIdx.x & 31;
  unsigned node = blockIdx.x * (blockDim.x >> 5) + (threadIdx.x >> 5);
  if (node >= n) return;
  size_t NK = (size_t)n * K;
  size_t base = (size_t)node * K;
  int k0 = lane, k1 = lane + 32;

  float z0 = -3.0e38f, z1 = -3.0e38f;
  if (k0 < K) {
    float m = out[2 * NK + base + k0];
    float lv = out[3 * NK + base + k0];
    float va = expf(lv);
    z0 = m + sqrtf(va) * eps[base + k0];
    out[base + k0] = z0;
    out[4 * NK + base + k0] = va;
  }
  if (k1 < K) {
    float m = out[2 * NK + base + k1];
    float lv = out[3 * NK + base + k1];
    float va = expf(lv);
    z1 = m + sqrtf(va) * eps[base + k1];
    out[base + k1] = z1;
    out[4 * NK + base + k1] = va;
  }
  float mx = fmaxf(z0, z1);
#pragma unroll
  for (int o = 16; o > 0; o >>= 1) mx = fmaxf(mx, __shfl_xor(mx, o, 32));
  float e0 = (k0 < K) ? expf(z0 - mx) : 0.0f;
  float e1 = (k1 < K) ? expf(z1 - mx) : 0.0f;
  float s = e0 + e1;
#pragma unroll
  for (int o = 16; o > 0; o >>= 1) s += __shfl_xor(s, o, 32);
  float inv = 1.0f / s;
  if (k0 < K) out[NK + base + k0] = e0 * inv;
  if (k1 < K) out[NK + base + k1] = e1 * inv;
}

// ---------------------------------------------------------------------------
extern "C" void kernel_launch(void* const* d_in, const int* in_sizes, int n_in,
                              void* d_out, int out_size, void* d_ws, size_t ws_size,
                              hipStream_t stream) {
  const float*     x      = (const float*)d_in[0];
  const long long* ei     = (const long long*)d_in[1];
  const float*     eps    = (const float*)d_in[2];
  const float*     W_base = (const float*)d_in[3];
  const float*     b_base = (const float*)d_in[4];
  const float*     W1     = (const float*)d_in[5];
  const float*     b1     = (const float*)d_in[6];
  const float*     W2     = (const float*)d_in[7];
  const float*     b2     = (const float*)d_in[8];
  const float*     W_mu   = (const float*)d_in[9];
  const float*     b_mu   = (const float*)d_in[10];
  const float*     W_ls   = (const float*)d_in[11];
  const float*     b_ls   = (const float*)d_in[12];

  const int HID = in_sizes[4];              // 256
  const int K   = in_sizes[10];             // 50
  const int IN  = in_sizes[3] / HID;        // 512
  const unsigned N = (unsigned)(in_sizes[0] / IN);   // 50000 (16 | N)
  const unsigned E = (unsigned)(in_sizes[1] / 2);    // 800000
  const int KP = 64;                        // K padded to WMMA tile multiple

  float* out  = (float*)d_out;
  float* P    = (float*)d_ws;               // N*HID   (h / agg ping)
  float* Q    = P + (size_t)N * HID;        // N*HID   (hw pong, also N*KP)
  float* dinv = Q + (size_t)N * HID;        // N
  float* Wp   = dinv + N;                   // HID*KP

  const long long* src = ei;
  const long long* dst = ei + E;

  // --- symmetric GCN normalization ---
  k_fill<<<(N + 255) / 256, 256, 0, stream>>>(dinv, 1.0f, N);  // self loop
  k_count_deg<<<(E + 255) / 256, 256, 0, stream>>>(dst, dinv, E);
  k_rsqrt_ip<<<(N + 255) / 256, 256, 0, stream>>>(dinv, N);

  auto gemm = [&](const float* A, const float* Wm, float* Cm, int fi, int fo) {
    size_t shmem = (size_t)16 * fi * sizeof(float);   // <= 32KB
    k_gemm_wmma_f32<<<N / 16, 256, shmem, stream>>>(A, Wm, Cm, fi, fo);
  };

  auto layer = [&](const float* A, const float* Wm, const float* bm, int fi) {
    gemm(A, Wm, Q, fi, HID);                                   // hw = A@W
    unsigned tot = N * (unsigned)HID;
    k_self_init<<<(tot + 255) / 256, 256, 0, stream>>>(Q, dinv, bm, P, N, HID, HID);
    k_edge_scatter256<<<E, 256, 0, stream>>>(Q, dinv, src, dst, P, E);
    k_softplus_ip<<<(tot + 255) / 256, 256, 0, stream>>>(P, tot);
  };

  layer(x, W_base, b_base, IN);   // P = softplus(gcn(x))
  layer(P, W1, b1, HID);          // A=P read only in gemm before P is rewritten
  layer(P, W2, b2, HID);

  size_t NK = (size_t)N * K;
  auto head = [&](const float* Wm, const float* bm, float* dest) {
    k_pad_w<<<(HID * KP + 255) / 256, 256, 0, stream>>>(Wm, Wp, HID, K, KP);
    gemm(P, Wp, Q, HID, KP);
    unsigned tot  = N * (unsigned)K;
    unsigned etot = E * (unsigned)K;
    k_self_init<<<(tot + 255) / 256, 256, 0, stream>>>(Q, dinv, bm, dest, N, KP, K);
    k_edge_scatter<<<(etot + 255) / 256, 256, 0, stream>>>(Q, dinv, src, dst, dest, E, KP, K);
  };
  head(W_mu, b_mu, out + 2 * NK);  // mu
  head(W_ls, b_ls, out + 3 * NK);  // logvar

  k_finalize<<<(N + 7) / 8, 256, 0, stream>>>(eps, out, N, K);
}